// Attention_54305566491075
// MI455X (gfx1250) — compile-verified
//
#include <hip/hip_runtime.h>

// ---------------------------------------------------------------------------
// Tacotron-style location-sensitive attention, fused for MI455X (gfx1250).
// fp32 WMMA (V_WMMA_F32_16X16X4_F32) for all GEMMs; flash-softmax so the
// 262 MB `memory` tensor is streamed from HBM only once for the energy GEMM
// (the context re-read hits L1/L2 for the just-touched tile).
// tanh -> native V_TANH_F32, exp -> native V_EXP_F32 to keep VALU lean.
// ---------------------------------------------------------------------------

typedef float v2f __attribute__((ext_vector_type(2)));
typedef float v8f __attribute__((ext_vector_type(8)));

#define B_    64
#define T_    2000
#define QD    1024
#define MD    512
#define AD    128
#define NF    32
#define KW    31
#define TPAD  2048
#define NCHUNK 32        // ceil(2000 / 64)

// workspace layout (float offsets)
#define PQ_OFF   0                         // 64*128      processed query
#define E_OFF    (PQ_OFF + B_*AD)          // 64*2048     raw energies
#define PM_OFF   (E_OFF + B_*TPAD)         // 64*32       per-chunk max
#define PS_OFF   (PM_OFF + B_*NCHUNK)      // 64*32       per-chunk expsum
#define PCTX_OFF (PS_OFF + B_*NCHUNK)      // 64*32*512   per-chunk context

#if defined(__has_builtin) && __has_builtin(__builtin_amdgcn_tanhf)
static __device__ inline float fast_tanh(float x) { return __builtin_amdgcn_tanhf(x); }
#else
static __device__ inline float fast_tanh(float x) { return tanhf(x); }
#endif
static __device__ inline float fast_exp(float x) { return __expf(x); }

static __device__ inline v8f wmma4(v2f a, v2f b, v8f c) {
  // D(16x16,f32) = A(16x4,f32) * B(4x16,f32) + C
  return __builtin_amdgcn_wmma_f32_16x16x4_f32(false, a, false, b,
                                               (short)0, c, false, false);
}

// ---------------------------------------------------------------------------
// K1: pq = tanh(hidden[64,1024] @ W_query[1024,128])  -> ws[PQ_OFF]
// one wave per 16x16 output tile; 256 K-steps of f32 WMMA
// ---------------------------------------------------------------------------
__global__ __launch_bounds__(32) void pq_kernel(const float* __restrict__ hid,
                                                const float* __restrict__ Wq,
                                                float* __restrict__ ws) {
  const int lane = threadIdx.x & 31;
  const int h = lane >> 4, nn = lane & 15;
  const int mt = blockIdx.x & 3, nt = blockIdx.x >> 2;
  const int m0 = mt * 16, n0 = nt * 16;

  v8f acc = {};
  const float* arow = hid + (size_t)(m0 + nn) * QD;   // A: M = lane&15
  #pragma unroll 4
  for (int ks = 0; ks < QD / 4; ++ks) {
    const int k0 = ks * 4 + 2 * h;                    // A/B frag K rows
    v2f a = *(const v2f*)(arow + k0);
    v2f b; b.x = Wq[(size_t)k0 * AD + n0 + nn];
           b.y = Wq[(size_t)(k0 + 1) * AD + n0 + nn];
    acc = wmma4(a, b, acc);
  }
  float* pq = ws + PQ_OFF;
  #pragma unroll
  for (int r = 0; r < 8; ++r) {
    const int row = m0 + r + 8 * h;                   // D: rows r / r+8
    pq[row * AD + n0 + nn] = fast_tanh(acc[r]);
  }
}

// ---------------------------------------------------------------------------
// K2: per (batch, 64-row T-chunk): conv -> ploc, pm GEMM, energies,
//     flash-softmax partials (m, s, ctx[512]) -> workspace
// 128 threads = 4 waves, one 16-row tile per wave.
// ---------------------------------------------------------------------------
__global__ __launch_bounds__(128) void attn_main_kernel(
    const float* __restrict__ memory,   // [B,T,MD]
    const float* __restrict__ awc,      // [B,2,T]
    const float* __restrict__ Wm,       // [MD,AD]
    const float* __restrict__ Wv,       // [AD]
    const float* __restrict__ Wc,       // [KW,2,NF]
    const float* __restrict__ Wl,       // [NF,AD]
    float* __restrict__ ws) {
  __shared__ float sWc[KW * 2 * NF];    // 7.9 KB
  __shared__ float sWl[NF * AD];        // 16 KB
  __shared__ float sPq[AD];
  __shared__ float sWv[AD];
  __shared__ float sEterm[4 * 16 * AD]; // 32 KB  tanh(ploc)+pq per wave tile
  __shared__ float sE16[4 * 16];
  __shared__ float sMw[4], sSw[4];
  __shared__ float sCtx[4 * MD];        // 8 KB

  const int tid = threadIdx.x;
  const int wave = tid >> 5, lane = tid & 31;
  const int h = lane >> 4, nn = lane & 15;
  const int chunk = blockIdx.x, b = blockIdx.y;
  const int t0 = chunk * 64 + wave * 16;

  for (int i = tid; i < KW * 2 * NF; i += 128) sWc[i] = Wc[i];
  for (int i = tid; i < NF * AD; i += 128)     sWl[i] = Wl[i];
  if (tid < AD) { sPq[tid] = ws[PQ_OFF + b * AD + tid]; sWv[tid] = Wv[tid]; }
  __syncthreads();

  // ---- Conv1D (SAME, no bias) straight into WMMA A-fragment layout:
  //      lane (h, m=nn) holds conv[m][{4j+2h, 4j+2h+1}]  (K = filter dim)
  const int tm = t0 + nn;
  v2f af8[8];
  #pragma unroll
  for (int j = 0; j < 8; ++j) af8[j] = (v2f){0.0f, 0.0f};
  const float* aw0 = awc + (size_t)(b * 2 + 0) * T_;
  const float* aw1 = awc + (size_t)(b * 2 + 1) * T_;
  for (int k = 0; k < KW; ++k) {
    const int it = tm + k - 15;
    if (it >= 0 && it < T_) {
      const float x0 = aw0[it], x1 = aw1[it];
      const float* w0 = &sWc[(k * 2 + 0) * NF];
      const float* w1 = &sWc[(k * 2 + 1) * NF];
      #pragma unroll
      for (int j = 0; j < 8; ++j) {
        const int f = 4 * j + 2 * h;
        af8[j].x += x0 * w0[f]     + x1 * w1[f];
        af8[j].y += x0 * w0[f + 1] + x1 * w1[f + 1];
      }
    }
  }

  // ---- ploc = tanh(conv @ W_loc); stash tanh(ploc)+pq in LDS
  #pragma unroll
  for (int nt = 0; nt < 8; ++nt) {
    const int n0 = nt * 16;
    v8f p = {};
    #pragma unroll
    for (int j = 0; j < 8; ++j) {
      const int k0 = 4 * j + 2 * h;
      v2f bb; bb.x = sWl[k0 * AD + n0 + nn];
              bb.y = sWl[(k0 + 1) * AD + n0 + nn];
      p = wmma4(af8[j], bb, p);
    }
    #pragma unroll
    for (int r = 0; r < 8; ++r)
      sEterm[wave * 16 * AD + (r + 8 * h) * AD + n0 + nn] =
          fast_tanh(p[r]) + sPq[n0 + nn];
  }

  // ---- pm GEMM: memory[16,512] @ W_memory[512,128], K tiled by 64
  const int tload = (tm < T_) ? tm : (T_ - 1);
  const float* mrow = memory + ((size_t)b * T_ + tload) * MD;
  v8f acc[8];
  #pragma unroll
  for (int nt = 0; nt < 8; ++nt) acc[nt] = (v8f){};
  #pragma unroll 1
  for (int kb = 0; kb < 8; ++kb) {
    const int kbase = kb * 64;
    v2f afr[16];
    #pragma unroll
    for (int ks = 0; ks < 16; ++ks)
      afr[ks] = *(const v2f*)(mrow + kbase + ks * 4 + 2 * h);
    if (kb < 7) __builtin_prefetch(mrow + kbase + 64, 0, 3);
    #pragma unroll
    for (int nt = 0; nt < 8; ++nt) {
      const int n0 = nt * 16;
      v8f a = acc[nt];
      #pragma unroll
      for (int ks = 0; ks < 16; ++ks) {
        const int k0 = kbase + ks * 4 + 2 * h;
        v2f bb; bb.x = Wm[(size_t)k0 * AD + n0 + nn];
                bb.y = Wm[(size_t)(k0 + 1) * AD + n0 + nn];
        a = wmma4(afr[ks], bb, a);
      }
      acc[nt] = a;
    }
  }

  // ---- energies: tanh(pq+ploc+pm) . W_v  (16-lane butterfly reduction)
  float ep[8];
  #pragma unroll
  for (int r = 0; r < 8; ++r) ep[r] = 0.0f;
  #pragma unroll
  for (int nt = 0; nt < 8; ++nt) {
    const int n0 = nt * 16;
    const float wv = sWv[n0 + nn];
    #pragma unroll
    for (int r = 0; r < 8; ++r) {
      float s = fast_tanh(acc[nt][r]) +
                sEterm[wave * 16 * AD + (r + 8 * h) * AD + n0 + nn];
      float u = fast_tanh(s) * wv;
      u += __shfl_xor(u, 1);
      u += __shfl_xor(u, 2);
      u += __shfl_xor(u, 4);
      u += __shfl_xor(u, 8);
      ep[r] += u;
    }
  }

  float* wsE = ws + E_OFF + (size_t)b * TPAD;
  #pragma unroll
  for (int r = 0; r < 8; ++r) {
    const int row = r + 8 * h;
    const int t = t0 + row;
    const float e = (t < T_) ? ep[r] : -1e30f;
    ep[r] = e;
    if (nn == r) {                 // unique writer per (row, half)
      if (t < T_) wsE[t] = e;
      sE16[wave * 16 + row] = e;
    }
  }

  // ---- per-wave flash stats
  float mh = ep[0];
  #pragma unroll
  for (int r = 1; r < 8; ++r) mh = fmaxf(mh, ep[r]);
  const float mw = fmaxf(mh, __shfl_xor(mh, 16));
  float sh = 0.0f;
  #pragma unroll
  for (int r = 0; r < 8; ++r) {
    const int t = t0 + r + 8 * h;
    sh += (t < T_) ? fast_exp(ep[r] - mw) : 0.0f;
  }
  const float sw = sh + __shfl_xor(sh, 16);
  __syncthreads();

  // ---- per-wave partial context: lane owns 16 contiguous columns
  {
    const int c0 = lane * 16;
    float a[16];
    #pragma unroll
    for (int i = 0; i < 16; ++i) a[i] = 0.0f;
    #pragma unroll 1
    for (int row = 0; row < 16; ++row) {
      const int t = t0 + row;
      if (t < T_) {
        const float w = fast_exp(sE16[wave * 16 + row] - mw);
        const float* mp = memory + ((size_t)b * T_ + t) * MD + c0;
        #pragma unroll
        for (int i = 0; i < 16; i += 4) {
          const float4 v = *(const float4*)(mp + i);
          a[i + 0] += w * v.x; a[i + 1] += w * v.y;
          a[i + 2] += w * v.z; a[i + 3] += w * v.w;
        }
      }
    }
    #pragma unroll
    for (int i = 0; i < 16; ++i) sCtx[wave * MD + c0 + i] = a[i];
    if (lane == 0) { sMw[wave] = mw; sSw[wave] = sw; }
  }
  __syncthreads();

  // ---- combine 4 waves -> one chunk partial in ws
  const float M = fmaxf(fmaxf(sMw[0], sMw[1]), fmaxf(sMw[2], sMw[3]));
  const float f0 = fast_exp(sMw[0] - M), f1 = fast_exp(sMw[1] - M);
  const float f2 = fast_exp(sMw[2] - M), f3 = fast_exp(sMw[3] - M);
  const float S = sSw[0] * f0 + sSw[1] * f1 + sSw[2] * f2 + sSw[3] * f3;
  const int pidx = b * NCHUNK + chunk;
  {
    const int c = tid * 4;                     // 128 threads * 4 = 512 cols
    float4 s4;
    s4.x = sCtx[c+0] * f0 + sCtx[MD+c+0] * f1 + sCtx[2*MD+c+0] * f2 + sCtx[3*MD+c+0] * f3;
    s4.y = sCtx[c+1] * f0 + sCtx[MD+c+1] * f1 + sCtx[2*MD+c+1] * f2 + sCtx[3*MD+c+1] * f3;
    s4.z = sCtx[c+2] * f0 + sCtx[MD+c+2] * f1 + sCtx[2*MD+c+2] * f2 + sCtx[3*MD+c+2] * f3;
    s4.w = sCtx[c+3] * f0 + sCtx[MD+c+3] * f1 + sCtx[2*MD+c+3] * f2 + sCtx[3*MD+c+3] * f3;
    *(float4*)(ws + PCTX_OFF + (size_t)pidx * MD + c) = s4;
  }
  if (tid == 0) { ws[PM_OFF + pidx] = M; ws[PS_OFF + pidx] = S; }
}

// ---------------------------------------------------------------------------
// K3: per-batch combine of 32 chunk partials -> context + softmax weights
// ---------------------------------------------------------------------------
__global__ __launch_bounds__(256) void combine_kernel(const float* __restrict__ ws,
                                                      float* __restrict__ out) {
  __shared__ float sm[NCHUNK], ss[NCHUNK];
  __shared__ float sMS[2];
  const int b = blockIdx.x, tid = threadIdx.x;
  if (tid < NCHUNK) {
    sm[tid] = ws[PM_OFF + b * NCHUNK + tid];
    ss[tid] = ws[PS_OFF + b * NCHUNK + tid];
  }
  __syncthreads();
  if (tid == 0) {
    float M = sm[0];
    for (int i = 1; i < NCHUNK; ++i) M = fmaxf(M, sm[i]);
    float S = 0.0f;
    for (int i = 0; i < NCHUNK; ++i) S += ss[i] * fast_exp(sm[i] - M);
    sMS[0] = M; sMS[1] = S;
  }
  __syncthreads();
  const float M = sMS[0], inv = 1.0f / sMS[1];

  // attention_context [B, 512]
  {
    const int c = tid * 2;                    // 256 threads * 2 = 512 cols
    float a0 = 0.0f, a1 = 0.0f;
    for (int i = 0; i < NCHUNK; ++i) {
      const float f = fast_exp(sm[i] - M);
      const float* p = ws + PCTX_OFF + (size_t)(b * NCHUNK + i) * MD + c;
      a0 += f * p[0]; a1 += f * p[1];
    }
    out[b * MD + c]     = a0 * inv;
    out[b * MD + c + 1] = a1 * inv;
  }
  // attention_weights [B, 2000]
  const float* e = ws + E_OFF + (size_t)b * TPAD;
  float* wout = out + B_ * MD + (size_t)b * T_;
  for (int t = tid; t < T_; t += 256)
    wout[t] = fast_exp(e[t] - M) * inv;
}

// ---------------------------------------------------------------------------
extern "C" void kernel_launch(void* const* d_in, const int* in_sizes, int n_in,
                              void* d_out, int out_size, void* d_ws, size_t ws_size,
                              hipStream_t stream) {
  const float* hid    = (const float*)d_in[0];  // [64,1024]
  const float* memory = (const float*)d_in[1];  // [64,2000,512]
  const float* awc    = (const float*)d_in[2];  // [64,2,2000]
  const float* Wq     = (const float*)d_in[3];  // [1024,128]
  const float* Wm     = (const float*)d_in[4];  // [512,128]
  const float* Wv     = (const float*)d_in[5];  // [128,1]
  const float* Wc     = (const float*)d_in[6];  // [31,2,32]
  const float* Wl     = (const float*)d_in[7];  // [32,128]
  float* ws  = (float*)d_ws;
  float* out = (float*)d_out;

  pq_kernel<<<32, 32, 0, stream>>>(hid, Wq, ws);
  attn_main_kernel<<<dim3(NCHUNK, B_), 128, 0, stream>>>(memory, awc, Wm, Wv, Wc, Wl, ws);
  combine_kernel<<<B_, 256, 0, stream>>>(ws, out);
}